// GNNRecommender_86406152060987
// MI455X (gfx1250) — compile-verified
//
#include <hip/hip_runtime.h>
#include <hip/hip_bf16.h>
#include <math.h>

// ---------------------------------------------------------------------------
// GNN recommender (3-layer bipartite GATv2 + MLP + BPR loss) for gfx1250.
// Dense GEMMs use the CDNA5 fp32 matrix pipe: V_WMMA_F32_16X16X4_F32
// (keeps fp32 reference numerics while using WMMA instead of VALU FMAs).
// ---------------------------------------------------------------------------

typedef float v2f __attribute__((ext_vector_type(2)));
typedef float v8f __attribute__((ext_vector_type(8)));

#define NEG_SLOPE 0.2f
#define LN_EPS 1e-5f

__device__ __forceinline__ float wave_sum(float v) {
#pragma unroll
  for (int o = 16; o > 0; o >>= 1) v += __shfl_xor(v, o, 32);
  return v;
}

__device__ __forceinline__ void atomicMaxF(float* addr, float val) {
  unsigned int* ua = (unsigned int*)addr;
  unsigned int cur = __float_as_uint(*addr);
  while (__uint_as_float(cur) < val) {
    unsigned int prev = atomicCAS(ua, cur, __float_as_uint(val));
    if (prev == cur) break;
    cur = prev;
  }
}

// ---------------------------------------------------------------------------
// WMMA fp32 GEMM:  C[M,N] = act(A[M,K] @ W[K,N] + bias[N])
// One wave computes a 16-row strip across the full N (NT = N/16 tiles kept in
// registers).  M%16==0, K%4==0, N==NT*16.
// A frag (16x4 f32): lane l<16 holds A[m0+l][k+0..1], lane l+16 holds k+2..3.
// B frag (4x16 f32): lane l<16 holds W[k+0..1][n0+l], lane l+16 holds k+2..3.
// C/D frag: vgpr r, lanes 0-15 -> row m0+r, lanes 16-31 -> row m0+8+r.
// ---------------------------------------------------------------------------
template <int NT>
__global__ __launch_bounds__(256) void k_gemm(const float* __restrict__ A,
                                              const float* __restrict__ W,
                                              const float* __restrict__ bias,
                                              float* __restrict__ C,
                                              int M, int K, int act) {
  const int N = NT * 16;
  int wid = (blockIdx.x * blockDim.x + threadIdx.x) >> 5;
  if (wid * 16 >= M) return;                      // wave-uniform exit
  int lane = threadIdx.x & 31;
  int lh = lane & 15;
  int hi = lane >> 4;                             // 0 or 1

  v8f acc[NT] = {};
  const float* arow = A + (size_t)(wid * 16 + lh) * K;

  for (int k = 0; k < K; k += 4) {
    v2f a;
    a.x = arow[k + 2 * hi];
    a.y = arow[k + 2 * hi + 1];
    const float* wr0 = W + (size_t)(k + 2 * hi) * N + lh;
    const float* wr1 = wr0 + N;
#pragma unroll
    for (int nt = 0; nt < NT; ++nt) {
      v2f b;
      b.x = wr0[nt * 16];
      b.y = wr1[nt * 16];
      acc[nt] = __builtin_amdgcn_wmma_f32_16x16x4_f32(
          /*neg_a=*/false, a, /*neg_b=*/false, b,
          /*c_mod=*/(short)0, acc[nt], /*reuse_a=*/false, /*reuse_b=*/false);
    }
  }

#pragma unroll
  for (int nt = 0; nt < NT; ++nt) {
    int ncol = nt * 16 + lh;
    float bv = bias ? bias[ncol] : 0.0f;
    size_t base = (size_t)(wid * 16 + hi * 8) * N + ncol;
#pragma unroll
    for (int r = 0; r < 8; ++r) {
      float v = acc[nt][r] + bv;
      if (act) v = fmaxf(v, 0.0f);
      C[base + (size_t)r * N] = v;
    }
  }
}

// ---------------------------------------------------------------------------
// Elementwise helpers
// ---------------------------------------------------------------------------
__global__ void k_fill(float* __restrict__ p, float v, size_t n) {
  size_t i = (size_t)blockIdx.x * blockDim.x + threadIdx.x;
  if (i < n) p[i] = v;
}

// replace -inf segment maxima with 0 (nodes with no in-edges) and zero sums
__global__ void k_fixm_zeros(float* __restrict__ m, float* __restrict__ s, int n) {
  int i = blockIdx.x * blockDim.x + threadIdx.x;
  if (i < n) {
    if (m[i] < -3.0e38f) m[i] = 0.0f;
    s[i] = 0.0f;
  }
}

// ---------------------------------------------------------------------------
// GATv2 edge pass 1: per edge, per head logits + atomic segment max.
// One wave per edge; H=4 heads, C channels per head, d = 4*C.
// ---------------------------------------------------------------------------
__global__ void k_edge_logits(const float* __restrict__ XL, const float* __restrict__ XR,
                              const int* __restrict__ src, const int* __restrict__ dst,
                              const float* __restrict__ att,       // [4*C]
                              float* __restrict__ logits,          // [E*4]
                              float* __restrict__ m,               // [n_dst*4]
                              int E, int d, int C) {
  int e = (blockIdx.x * blockDim.x + threadIdx.x) >> 5;
  if (e >= E) return;
  int lane = threadIdx.x & 31;
  int sn = src[e], tn = dst[e];
  const float* xl = XL + (size_t)sn * d;
  const float* xr = XR + (size_t)tn * d;
#pragma unroll
  for (int h = 0; h < 4; ++h) {
    float a = 0.0f;
    for (int c = lane; c < C; c += 32) {
      int ch = h * C + c;
      float v = xl[ch] + xr[ch];
      v = v > 0.0f ? v : NEG_SLOPE * v;
      a += v * att[ch];
    }
    float r = wave_sum(a);
    if (lane == 0) {
      logits[(size_t)e * 4 + h] = r;
      atomicMaxF(&m[(size_t)tn * 4 + h], r);
    }
  }
}

// edge pass 2: a = exp(logit - m[dst]); segment sum (one thread per edge*head)
__global__ void k_edge_expsum(const float* __restrict__ logits,
                              const float* __restrict__ m,
                              const int* __restrict__ dst,
                              float* __restrict__ s, int E) {
  int i = blockIdx.x * blockDim.x + threadIdx.x;
  if (i >= E * 4) return;
  int e = i >> 2, h = i & 3;
  int tn = dst[e];
  float a = __expf(logits[i] - m[(size_t)tn * 4 + h]);
  atomicAdd(&s[(size_t)tn * 4 + h], a);
}

// edge pass 3: out[dst] += alpha * xl[src]   (one wave per edge)
__global__ void k_edge_scatter(const float* __restrict__ XL,
                               const float* __restrict__ logits,
                               const float* __restrict__ m,
                               const float* __restrict__ s,
                               const int* __restrict__ src,
                               const int* __restrict__ dst,
                               float* __restrict__ out,
                               int E, int d, int C) {
  int e = (blockIdx.x * blockDim.x + threadIdx.x) >> 5;
  if (e >= E) return;
  int lane = threadIdx.x & 31;
  int sn = src[e], tn = dst[e];
  const float* xl = XL + (size_t)sn * d;
  float* orow = out + (size_t)tn * d;
  for (int ch = lane; ch < d; ch += 32) {
    int h = ch / C;
    float lg = logits[(size_t)e * 4 + h];
    float mm = m[(size_t)tn * 4 + h];
    float ss = s[(size_t)tn * 4 + h];
    float alpha = __expf(lg - mm) / (ss + 1e-16f);
    atomicAdd(&orow[ch], alpha * xl[ch]);
  }
}

// ---------------------------------------------------------------------------
// LayerNorm(+gat bias)(+residual)(+ELU), in place.  One wave per row, d<=128.
// ---------------------------------------------------------------------------
__global__ void k_ln(float* __restrict__ Z,
                     const float* __restrict__ gbias,
                     const float* __restrict__ g, const float* __restrict__ b,
                     const float* __restrict__ resid,   // may be null
                     int n, int d, int do_elu) {
  int row = (blockIdx.x * blockDim.x + threadIdx.x) >> 5;
  if (row >= n) return;
  int lane = threadIdx.x & 31;
  float vals[4];
  float sum = 0.0f;
#pragma unroll
  for (int j = 0; j < 4; ++j) {
    int ch = lane + 32 * j;
    float v = (ch < d) ? (Z[(size_t)row * d + ch] + gbias[ch]) : 0.0f;
    vals[j] = v;
    sum += v;
  }
  sum = wave_sum(sum);
  float mu = sum / (float)d;
  float vs = 0.0f;
#pragma unroll
  for (int j = 0; j < 4; ++j) {
    int ch = lane + 32 * j;
    if (ch < d) { float dv = vals[j] - mu; vs += dv * dv; }
  }
  vs = wave_sum(vs) / (float)d;
  float rstd = rsqrtf(vs + LN_EPS);
#pragma unroll
  for (int j = 0; j < 4; ++j) {
    int ch = lane + 32 * j;
    if (ch < d) {
      float z = (vals[j] - mu) * rstd * g[ch] + b[ch];
      if (resid) z += resid[(size_t)row * d + ch];
      if (do_elu) z = z > 0.0f ? z : (__expf(z) - 1.0f);
      Z[(size_t)row * d + ch] = z;
    }
  }
}

// ---------------------------------------------------------------------------
// Batch gather + concat:  X[2B, 128]  rows 0..B-1 = [zU, zPos], B..2B-1 = [zU, zNeg]
// ---------------------------------------------------------------------------
__global__ void k_gather(const float* __restrict__ hU, const float* __restrict__ hI,
                         const int* __restrict__ uidx, const int* __restrict__ pidx,
                         const int* __restrict__ nidx, float* __restrict__ X, int B) {
  int i = blockIdx.x * blockDim.x + threadIdx.x;
  int total = 2 * B * 128;
  if (i >= total) return;
  int r = i >> 7, c = i & 127;
  int rb = (r >= B) ? (r - B) : r;
  if (c < 64) {
    X[i] = hU[(size_t)uidx[rb] * 64 + c];
  } else {
    int it = (r >= B) ? nidx[rb] : pidx[rb];
    X[i] = hI[(size_t)it * 64 + (c - 64)];
  }
}

// final dot with W3 + BPR loss reduction into out[0] (pre-zeroed)
__global__ void k_score_loss(const float* __restrict__ A2,   // [2B, 64]
                             const float* __restrict__ W3,   // [64]
                             const float* __restrict__ b3,   // [1]
                             float* __restrict__ out, int B) {
  int i = blockIdx.x * blockDim.x + threadIdx.x;
  if (i >= B) return;
  float p = 0.0f, q = 0.0f;
#pragma unroll 4
  for (int c = 0; c < 64; ++c) {
    float w = W3[c];
    p += A2[(size_t)i * 64 + c] * w;
    q += A2[(size_t)(i + B) * 64 + c] * w;
  }
  float x = (p + b3[0]) - (q + b3[0]);
  float ls = fminf(x, 0.0f) - log1pf(__expf(-fabsf(x)));   // log_sigmoid(x)
  atomicAdd(out, -ls / (float)B);
}

// ---------------------------------------------------------------------------
// Host orchestration
// ---------------------------------------------------------------------------
static void launch_gemm(const float* A, const float* W, const float* b, float* C,
                        int M, int N, int K, int act, hipStream_t s) {
  dim3 bl(256), gr((M / 16 + 7) / 8);
  if (N == 128)
    hipLaunchKernelGGL((k_gemm<8>), gr, bl, 0, s, A, W, b, C, M, K, act);
  else
    hipLaunchKernelGGL((k_gemm<4>), gr, bl, 0, s, A, W, b, C, M, K, act);
}

static inline dim3 g1(size_t n, int bs) { return dim3((unsigned)((n + bs - 1) / bs)); }

extern "C" void kernel_launch(void* const* d_in, const int* in_sizes, int n_in,
                              void* d_out, int out_size, void* d_ws, size_t ws_size,
                              hipStream_t stream) {
  if (n_in < 65) return;   // unexpected flattening; bail deterministically

  const int IN_DIM = 64, HID = 128;
  const int NU = in_sizes[0] / IN_DIM;       // 100000
  const int NI = in_sizes[1] / IN_DIM;       // 50000

  // Assumed flatten order: top-level dict insertion order; "params" flattened
  // JAX-style (dict keys sorted):
  //   per layer L (base = 2 + 16L):
  //     +0..5   iu: Wl, Wr, att, bias, bl, br
  //     +6,+7   ln_item: gamma, beta
  //     +8,+9   ln_user: gamma, beta
  //     +10..15 ui: Wl, Wr, att, bias, bl, br
  //   pred (50..55): W1, W2, W3, b1, b2, b3
  //   56..59: proj_item_W, proj_item_b, proj_user_W, proj_user_b
  const int PBASE = 2, PRED = PBASE + 48;
  const int P_ITEM_W = PRED + 6, P_ITEM_B = PRED + 7;
  const int P_USER_W = PRED + 8, P_USER_B = PRED + 9;
  const int IDX_EUI = 60, IDX_EIU = 61, IDX_U = 62, IDX_P = 63, IDX_N = 64;

  auto F = [&](int i) { return (const float*)d_in[i]; };
  const int E = in_sizes[IDX_EUI] / 2;       // 1000000
  const int B = in_sizes[IDX_U];             // 16384
  const int* src_ui = (const int*)d_in[IDX_EUI];
  const int* dst_ui = src_ui + E;
  const int* src_iu = (const int*)d_in[IDX_EIU];
  const int* dst_iu = src_iu + E;

  // -------- workspace carve-up (fp32) --------
  float* w = (float*)d_ws;
  size_t off = 0;
  auto alloc = [&](size_t n) { float* p = w + off; off += n; return p; };
  float* hU  = alloc((size_t)NU * HID);
  float* hI  = alloc((size_t)NI * HID);
  float* hnU = alloc((size_t)NU * HID);
  float* hnI = alloc((size_t)NI * HID);
  float* XL  = alloc((size_t)NU * HID);      // source transform (max size)
  float* XR  = alloc((size_t)NU * HID);      // dest transform (max size)
  float* LG  = alloc((size_t)E * 4);         // edge logits
  float* Mb  = alloc((size_t)NU * 4);        // segment max
  float* Sb  = alloc((size_t)NU * 4);        // segment sum
  float* Xc  = alloc((size_t)2 * B * 128);   // concat batch
  float* A1  = alloc((size_t)2 * B * 128);
  float* A2  = alloc((size_t)2 * B * 64);
  (void)ws_size;

  const int BS = 256;

  // -------- input projections --------
  launch_gemm(F(0), F(P_USER_W), F(P_USER_B), hU, NU, HID, IN_DIM, 0, stream);
  launch_gemm(F(1), F(P_ITEM_W), F(P_ITEM_B), hI, NI, HID, IN_DIM, 0, stream);

  // -------- GAT layers --------
  float* hu = hU; float* hi_ = hI; float* hnu = hnU; float* hni = hnI;
  int din = HID;
  for (int L = 0; L < 3; ++L) {
    const int dout = (L == 2) ? 64 : 128;
    const int C = dout / 4;
    const int base = PBASE + 16 * L;

    // ---- ui conv: user -> item, writes hni ----
    launch_gemm(hu,  F(base + 10), F(base + 14), XL, NU, dout, din, 0, stream); // xl=user@Wl+bl
    launch_gemm(hi_, F(base + 11), F(base + 15), XR, NI, dout, din, 0, stream); // xr=item@Wr+br
    hipLaunchKernelGGL(k_fill, g1((size_t)NI * 4, BS), dim3(BS), 0, stream,
                       Mb, -__builtin_huge_valf(), (size_t)NI * 4);
    hipLaunchKernelGGL(k_edge_logits, g1((size_t)E * 32, BS), dim3(BS), 0, stream,
                       XL, XR, src_ui, dst_ui, F(base + 12), LG, Mb, E, dout, C);
    hipLaunchKernelGGL(k_fixm_zeros, g1((size_t)NI * 4, BS), dim3(BS), 0, stream,
                       Mb, Sb, NI * 4);
    hipLaunchKernelGGL(k_edge_expsum, g1((size_t)E * 4, BS), dim3(BS), 0, stream,
                       LG, Mb, dst_ui, Sb, E);
    hipLaunchKernelGGL(k_fill, g1((size_t)NI * dout, BS), dim3(BS), 0, stream,
                       hni, 0.0f, (size_t)NI * dout);
    hipLaunchKernelGGL(k_edge_scatter, g1((size_t)E * 32, BS), dim3(BS), 0, stream,
                       XL, LG, Mb, Sb, src_ui, dst_ui, hni, E, dout, C);
    hipLaunchKernelGGL(k_ln, g1((size_t)NI * 32, BS), dim3(BS), 0, stream,
                       hni, F(base + 13), F(base + 6), F(base + 7),
                       (dout == din) ? hi_ : nullptr, NI, dout, (L < 2) ? 1 : 0);

    // ---- iu conv: item -> user, writes hnu ----
    launch_gemm(hi_, F(base + 0), F(base + 4), XL, NI, dout, din, 0, stream);  // xl=item@Wl+bl
    launch_gemm(hu,  F(base + 1), F(base + 5), XR, NU, dout, din, 0, stream);  // xr=user@Wr+br
    hipLaunchKernelGGL(k_fill, g1((size_t)NU * 4, BS), dim3(BS), 0, stream,
                       Mb, -__builtin_huge_valf(), (size_t)NU * 4);
    hipLaunchKernelGGL(k_edge_logits, g1((size_t)E * 32, BS), dim3(BS), 0, stream,
                       XL, XR, src_iu, dst_iu, F(base + 2), LG, Mb, E, dout, C);
    hipLaunchKernelGGL(k_fixm_zeros, g1((size_t)NU * 4, BS), dim3(BS), 0, stream,
                       Mb, Sb, NU * 4);
    hipLaunchKernelGGL(k_edge_expsum, g1((size_t)E * 4, BS), dim3(BS), 0, stream,
                       LG, Mb, dst_iu, Sb, E);
    hipLaunchKernelGGL(k_fill, g1((size_t)NU * dout, BS), dim3(BS), 0, stream,
                       hnu, 0.0f, (size_t)NU * dout);
    hipLaunchKernelGGL(k_edge_scatter, g1((size_t)E * 32, BS), dim3(BS), 0, stream,
                       XL, LG, Mb, Sb, src_iu, dst_iu, hnu, E, dout, C);
    hipLaunchKernelGGL(k_ln, g1((size_t)NU * 32, BS), dim3(BS), 0, stream,
                       hnu, F(base + 3), F(base + 8), F(base + 9),
                       (dout == din) ? hu : nullptr, NU, dout, (L < 2) ? 1 : 0);

    // swap old/new
    float* t;
    t = hu; hu = hnu; hnu = t;
    t = hi_; hi_ = hni; hni = t;
    din = dout;
  }

  // -------- prediction MLP + BPR loss --------
  hipLaunchKernelGGL(k_gather, g1((size_t)2 * B * 128, BS), dim3(BS), 0, stream,
                     hu, hi_, (const int*)d_in[IDX_U], (const int*)d_in[IDX_P],
                     (const int*)d_in[IDX_N], Xc, B);
  launch_gemm(Xc, F(PRED + 0), F(PRED + 3), A1, 2 * B, 128, 128, 1, stream);  // W1,b1,relu
  launch_gemm(A1, F(PRED + 1), F(PRED + 4), A2, 2 * B, 64, 128, 1, stream);   // W2,b2,relu
  hipLaunchKernelGGL(k_fill, dim3(1), dim3(32), 0, stream, (float*)d_out, 0.0f, (size_t)1);
  hipLaunchKernelGGL(k_score_loss, g1((size_t)B, BS), dim3(BS), 0, stream,
                     A2, F(PRED + 2), F(PRED + 5), (float*)d_out, B);
}